// DynaKeyMemoryCore_8358006358506
// MI455X (gfx1250) — compile-verified
//
#include <hip/hip_runtime.h>
#include <math.h>

// Problem constants (from reference): B=2,N=8,C=256,H=W=64,K=16, DT=1, TEMP=1
#define BN   16      // B*N
#define CC   256     // channels
#define HW   4096    // H*W
#define KK   16      // memory slots per (b,n)

typedef __attribute__((ext_vector_type(2))) float v2f;
typedef __attribute__((ext_vector_type(8))) float v8f;

// ---------------------------------------------------------------------------
// Kernel A: masked spatial pooling. One block per (bn,c) row of 4096 pixels.
// Fully coalesced streaming read of value (64MB) + mask (L2-resident).
// Produces wsum[bnc] = sum(value*mask), vsum[bnc] = sum(value).
// ---------------------------------------------------------------------------
__global__ __launch_bounds__(256) void pool_kernel(const float* __restrict__ value,
                                                   const float* __restrict__ mask,
                                                   float* __restrict__ wsum,
                                                   float* __restrict__ vsum) {
    const int bnc = blockIdx.x;          // bn*C + c
    const int bn  = bnc >> 8;
    const float* v = value + (size_t)bnc * HW;
    const float* m = mask  + (size_t)bn  * HW;

    float ws = 0.f, vs = 0.f;
    for (int i = threadIdx.x; i < HW; i += 256) {
        float vv = v[i];
        ws = fmaf(vv, m[i], ws);
        vs += vv;
    }
    // wave32 reduction then cross-wave via LDS
    for (int off = 16; off > 0; off >>= 1) {
        ws += __shfl_xor(ws, off, 32);
        vs += __shfl_xor(vs, off, 32);
    }
    __shared__ float rw[8], rv[8];
    const int wave = threadIdx.x >> 5, lane = threadIdx.x & 31;
    if (lane == 0) { rw[wave] = ws; rv[wave] = vs; }
    __syncthreads();
    if (threadIdx.x == 0) {
        float a = 0.f, b = 0.f;
        #pragma unroll
        for (int i = 0; i < 8; ++i) { a += rw[i]; b += rv[i]; }
        wsum[bnc] = a;
        vsum[bnc] = b;
    }
}

// ---------------------------------------------------------------------------
// Kernel B (single 256-thread block, 8 waves):
//   1) denom[bn] = sum(mask)          2) z[bn,c] (pooled w/ mean fallback)
//   3) d2 -> logits -> softmax weights w[bn,k]
//   4) delta[bn,c] = gate * sum_k w[bn,k]*vel[bn,k,c] via V_WMMA_F32_16X16X4_F32
//      as a real GEMM: A(16x256) block-diagonal in w, B(256x256) = velocities.
// ---------------------------------------------------------------------------
__global__ __launch_bounds__(256) void state_kernel(const float* __restrict__ mask,
                                                    const float* __restrict__ keys,
                                                    const float* __restrict__ vel,
                                                    const unsigned char* __restrict__ valid,
                                                    const float* __restrict__ gatep,
                                                    const float* __restrict__ wsum,
                                                    const float* __restrict__ vsum,
                                                    float* __restrict__ delta) {
    __shared__ float zs[BN * CC];     // 16 KB: pooled state z
    __shared__ float denom[BN];
    __shared__ float red[256];
    __shared__ float wlds[BN * KK];   // softmax weights

    const int t = threadIdx.x;

    // --- 1) denom per (b,n): 16 threads per bn ---
    {
        const int bn = t >> 4, sub = t & 15;
        float p = 0.f;
        for (int i = sub; i < HW; i += 16) p += mask[bn * HW + i];
        red[t] = p;
    }
    __syncthreads();
    if (t < BN) {
        float s = 0.f;
        #pragma unroll
        for (int j = 0; j < 16; ++j) s += red[t * 16 + j];
        denom[t] = s;
    }
    __syncthreads();

    // --- 2) z = pooled (or mean fallback) ---
    for (int idx = t; idx < BN * CC; idx += 256) {
        const float den = denom[idx >> 8];
        zs[idx] = (den > 1e-5f) ? (wsum[idx] / fmaxf(den, 1e-6f))
                                : (vsum[idx] * (1.0f / (float)HW));
    }
    __syncthreads();

    // --- 3) squared distances + masked softmax over K (TEMP=1) ---
    float d2 = 0.f;
    {
        const float* kp = keys + (size_t)t * CC;     // pair p = bn*16+k = t
        const float* zp = zs + (t >> 4) * CC;
        for (int c = 0; c < CC; ++c) {
            const float df = zp[c] - kp[c];
            d2 = fmaf(df, df, d2);
        }
    }
    const float logit = valid[t] ? (-d2) : -INFINITY;
    red[t] = logit;
    __syncthreads();
    const int row0 = t & ~15;
    float mx = -INFINITY;
    #pragma unroll
    for (int j = 0; j < 16; ++j) mx = fmaxf(mx, red[row0 + j]);
    const float e = (logit == -INFINITY) ? 0.f : __expf(logit - mx);
    __syncthreads();
    red[t] = e;
    __syncthreads();
    float sum = 0.f;
    #pragma unroll
    for (int j = 0; j < 16; ++j) sum += red[row0 + j];
    wlds[t] = (sum > 0.f) ? (e / sum) : 0.f;
    __syncthreads();

    // --- 4) WMMA GEMM: D(16x256) = Ablockdiag(16x256) x vel(256x256) ---
    // f32 16x16x4 A layout: lane L -> M = L&15, VGPR0/1 hold K = 2*(L>>4)+{0,1}
    // D layout: VGPR r -> M = r + 8*(L>>4), N = L&15
    const float gate = *gatep;
    const int wave = t >> 5, lane = t & 31;
    const int Mrow  = lane & 15;
    const int khalf = 2 * (lane >> 4);

    for (int tile = wave; tile < CC / 16; tile += 8) {     // 8 waves, 2 c-tiles each
        const int c0 = tile * 16;
        const int cc = c0 + (lane & 15);
        v8f acc = {};                                      // C = 0
        for (int kk0 = 0; kk0 < BN * KK; kk0 += 4) {
            const int kb = kk0 + khalf;                    // global K index (bn'*16+k)
            // Unconditional LDS reads (lane-uniform address -> broadcast read),
            // then select so the block-diagonal mask lowers to v_cndmask, not
            // exec-predicated ds_load.
            const float w0 = wlds[kb];
            const float w1 = wlds[kb + 1];
            const float a0 = ((kb >> 4) == Mrow) ? w0 : 0.f;
            const float a1 = (((kb + 1) >> 4) == Mrow) ? w1 : 0.f;
            // B: vel laid out [bn',k][c] row-major -> B[kb][n] = vel[kb*CC + c0+n]
            const float b0 = vel[(size_t)kb * CC + cc];
            const float b1 = vel[(size_t)(kb + 1) * CC + cc];
            v2f Af = {a0, a1};
            v2f Bf = {b0, b1};
            acc = __builtin_amdgcn_wmma_f32_16x16x4_f32(false, Af, false, Bf,
                                                        (short)0, acc, false, false);
        }
        const int mbase = 8 * (lane >> 4);
        #pragma unroll
        for (int r = 0; r < 8; ++r)
            delta[(mbase + r) * CC + cc] = gate * acc[r];  // gate * (z_next - z)
    }
}

// ---------------------------------------------------------------------------
// Kernel C: streaming readout out = value + delta[bnc], float4 vectorized.
// ---------------------------------------------------------------------------
__global__ __launch_bounds__(256) void readout_kernel(const float* __restrict__ value,
                                                      const float* __restrict__ delta,
                                                      float* __restrict__ out,
                                                      int nvec4) {
    const float4* __restrict__ v4 = (const float4*)value;
    float4* __restrict__ o4 = (float4*)out;
    const int stride = gridDim.x * blockDim.x;
    for (int i = blockIdx.x * blockDim.x + threadIdx.x; i < nvec4; i += stride) {
        __builtin_prefetch(&v4[i + stride], 0, 0);   // global_prefetch_b8
        const float4 v = v4[i];
        const float d = delta[i >> 10];              // 4096 floats = 1024 float4 per (bn,c)
        o4[i] = make_float4(v.x + d, v.y + d, v.z + d, v.w + d);
    }
}

// ---------------------------------------------------------------------------
extern "C" void kernel_launch(void* const* d_in, const int* in_sizes, int n_in,
                              void* d_out, int out_size, void* d_ws, size_t ws_size,
                              hipStream_t stream) {
    const float*         value = (const float*)d_in[0];
    /* d_in[1] key_feat and d_in[2] pixfeat are unused by the reference output */
    const float*         mask  = (const float*)d_in[3];
    const float*         keys  = (const float*)d_in[4];
    const float*         vel   = (const float*)d_in[5];
    const unsigned char* valid = (const unsigned char*)d_in[6];
    const float*         gate  = (const float*)d_in[7];
    float* out = (float*)d_out;

    float* ws   = (float*)d_ws;
    float* wsum = ws;                 // BN*CC floats
    float* vsum = ws + BN * CC;       // BN*CC floats
    float* delt = ws + 2 * BN * CC;   // BN*CC floats

    pool_kernel<<<BN * CC, 256, 0, stream>>>(value, mask, wsum, vsum);
    state_kernel<<<1, 256, 0, stream>>>(mask, keys, vel, valid, gate, wsum, vsum, delt);
    readout_kernel<<<4096, 256, 0, stream>>>(value, delt, out, (BN * CC * HW) / 4);
}